// Deformable1DFeatureAggregator_82437602279993
// MI455X (gfx1250) — compile-verified
//
#include <hip/hip_runtime.h>
#include <math.h>

typedef float v2f __attribute__((ext_vector_type(2)));
typedef float v8f __attribute__((ext_vector_type(8)));
typedef int   v4i __attribute__((ext_vector_type(4)));

#define B_   2
#define C_   256
#define H_   128
#define W_   256
#define HW_  32768
#define P_   9
#define G_   8
#define GP_  72
#define MTOT 65536

#ifndef __has_builtin
#define __has_builtin(x) 0
#endif

// CDNA5 async global->LDS staging (ASYNCcnt-tracked). Guarded: falls back to
// synchronous float4 staging if this toolchain doesn't declare the builtins.
// Probe result (round 2): builtin exists with signature
//   (v4i addrspace(1)*, v4i addrspace(3)*, imm offset, imm cpol)
#if __has_builtin(__builtin_amdgcn_global_load_async_to_lds_b128) && \
    __has_builtin(__builtin_amdgcn_s_wait_asynccnt)
#define USE_ASYNC 1
#define AS1_ __attribute__((address_space(1)))
#define AS3_ __attribute__((address_space(3)))
#define ASYNC_LOAD_B128(gptr, lptr)                                        \
  __builtin_amdgcn_global_load_async_to_lds_b128(                          \
      (AS1_ v4i*)(gptr), (AS3_ v4i*)(lptr), 0, 0)
#define ASYNC_WAIT() __builtin_amdgcn_s_wait_asynccnt(0)
#else
#define USE_ASYNC 0
#define ASYNC_WAIT()
#endif

// ---------------- K0: LayerNorm statistics (mean / rstd over C per pixel) ----------------
__global__ __launch_bounds__(256) void k0_stats(
    const float* __restrict__ f1, const float* __restrict__ f2,
    float* __restrict__ mean1, float* __restrict__ rstd1,
    float* __restrict__ mean2, float* __restrict__ rstd2)
{
  const float* f = blockIdx.y ? f2 : f1;
  float* mo = blockIdx.y ? mean2 : mean1;
  float* ro = blockIdx.y ? rstd2 : rstd1;
  const int tid = threadIdx.x;
  const int tx = tid & 31, ty = tid >> 5;
  const int m0 = blockIdx.x * 32;
  const int b = m0 / HW_, n0 = m0 % HW_;
  float s = 0.f, ss = 0.f;
  for (int c = ty; c < C_; c += 8) {
    float v = f[((size_t)(b * C_ + c)) * HW_ + n0 + tx];
    s += v; ss += v * v;
  }
  __shared__ float S[8][32], SS[8][32];
  S[ty][tx] = s; SS[ty][tx] = ss;
  __syncthreads();
  if (ty == 0) {
    float a = 0.f, aa = 0.f;
    for (int i = 0; i < 8; ++i) { a += S[i][tx]; aa += SS[i][tx]; }
    float mu = a * (1.f / C_);
    float var = aa * (1.f / C_) - mu * mu;
    mo[m0 + tx] = mu;
    ro[m0 + tx] = rsqrtf(var + 1e-5f);
  }
}

// ---------------- K1: value = LN(feats2) @ Wv + bv  (fp32 WMMA, async double-buffered) ----
__global__ __launch_bounds__(256) void k1_value(
    const float* __restrict__ feats2,
    const float* __restrict__ mean2, const float* __restrict__ rstd2,
    const float* __restrict__ g2, const float* __restrict__ be2,
    const float* __restrict__ Wv, const float* __restrict__ bv,
    float* __restrict__ value)
{
  __shared__ float Als[2][32 * 33];    // [buf][pixel][k]  stride 33
  __shared__ float Bls[2][32 * 132];   // [buf][k][n]      stride 132 (16B aligned rows)
  const int tid = threadIdx.x;
  const int m0 = blockIdx.x * 32;
  const int nb = blockIdx.y * 128;
  const int b = m0 / HW_, n_in = m0 % HW_;
  const int wid = tid >> 5, lane = tid & 31;
  const int wave_m = wid >> 2, wave_n = wid & 3;
  const int lane16 = lane & 15;
  const int koff = (lane < 16) ? 0 : 2;       // A/B K-pair layout: VGPR0={K0|K2}, VGPR1={K1|K3}
  const int m_lane = wave_m * 16 + lane16;

  const int lmm = tid & 31;   // pixel this thread stages
  const int lk  = tid >> 5;   // base k-row this thread stages
  const float mu = mean2[m0 + lmm];
  const float rs = rstd2[m0 + lmm];

  v8f acc0 = {}, acc1 = {};

  auto stageA = [&](int buf, int k0) {        // LN fused into A staging
    for (int i = 0; i < 4; ++i) {
      int kk = lk + 8 * i;
      float fv = feats2[((size_t)(b * C_ + k0 + kk)) * HW_ + n_in + lmm];
      Als[buf][lmm * 33 + kk] = (fv - mu) * rs * g2[k0 + kk] + be2[k0 + kk];
    }
  };
  auto stageB = [&](int buf, int k0) {
#if USE_ASYNC
    // one async b128 per wave per row: 32 lanes x 16B = one 512B row of Wv -> LDS
    for (int i = 0; i < 4; ++i) {
      int r = wid + 8 * i;
      const float* gsrc = Wv + (size_t)(k0 + r) * C_ + nb + lane * 4;
      float* ldst = &Bls[buf][r * 132 + lane * 4];
      ASYNC_LOAD_B128(gsrc, ldst);
    }
#else
    int cg = (tid & 31) * 4;
    int r0 = tid >> 5;
    for (int i = 0; i < 4; ++i) {
      int r = r0 + 8 * i;
      float4 w = *(const float4*)(Wv + (size_t)(k0 + r) * C_ + nb + cg);
      *(float4*)(&Bls[buf][r * 132 + cg]) = w;
    }
#endif
  };
  auto compute = [&](int buf) {
    const int ncol = wave_n * 32 + lane16;
    const float* A = &Als[buf][0];
    const float* Bm = &Bls[buf][0];
    for (int kk = 0; kk < 32; kk += 4) {
      v2f a;  a.x  = A[m_lane * 33 + kk + koff];        a.y  = A[m_lane * 33 + kk + koff + 1];
      v2f b0; b0.x = Bm[(kk + koff) * 132 + ncol];      b0.y = Bm[(kk + koff + 1) * 132 + ncol];
      v2f b1; b1.x = Bm[(kk + koff) * 132 + ncol + 16]; b1.y = Bm[(kk + koff + 1) * 132 + ncol + 16];
      acc0 = __builtin_amdgcn_wmma_f32_16x16x4_f32(false, a, false, b0, (short)0, acc0, false, false);
      acc1 = __builtin_amdgcn_wmma_f32_16x16x4_f32(false, a, false, b1, (short)0, acc1, false, false);
    }
  };

  stageA(0, 0);
  stageB(0, 0);
  ASYNC_WAIT();
  __syncthreads();
  for (int c = 0; c < 8; ++c) {
    const int cur = c & 1, nxt = cur ^ 1;
    if (c < 7) { stageA(nxt, (c + 1) * 32); stageB(nxt, (c + 1) * 32); }
    compute(cur);
    ASYNC_WAIT();
    __syncthreads();
  }

  const int c0 = nb + wave_n * 32 + lane16;
  const float bva = bv[c0], bvb = bv[c0 + 16];
  const int rbase = m0 + wave_m * 16 + ((lane < 16) ? 0 : 8);
  for (int v = 0; v < 8; ++v) {
    value[(size_t)(rbase + v) * C_ + c0]      = acc0[v] + bva;
    value[(size_t)(rbase + v) * C_ + c0 + 16] = acc1[v] + bvb;
  }
}

// ---------------- K2: LN(feats1) @ [Ww|Wk] -> softmax weights + key points ----------------
__global__ __launch_bounds__(256) void k2_wproj(
    const float* __restrict__ feats1,
    const float* __restrict__ mean1, const float* __restrict__ rstd1,
    const float* __restrict__ g1, const float* __restrict__ be1,
    const float* __restrict__ Ww, const float* __restrict__ bw,
    const float* __restrict__ Wk, const float* __restrict__ bk,
    const float* __restrict__ anchors,
    float* __restrict__ wts, float* __restrict__ kp_out)
{
  __shared__ float Als[32 * 257];   // full-K LN'd activations, 32 pixels
  __shared__ float Wls[32 * 97];    // K-chunk of packed [Ww(72) | Wk(9) | pad(15)]
  __shared__ float Lls[32 * 96];    // logits tile
  const int tid = threadIdx.x;
  const int m0 = blockIdx.x * 32;
  const int b = m0 / HW_, n_in = m0 % HW_;
  const int wid = tid >> 5, lane = tid & 31;
  const int lane16 = lane & 15;
  const int koff = (lane < 16) ? 0 : 2;
  {
    const int lmm = tid & 31;
    const float mu = mean1[m0 + lmm];
    const float rs = rstd1[m0 + lmm];
    for (int i = 0; i < 32; ++i) {
      int c = (tid >> 5) + 8 * i;
      float fv = feats1[((size_t)(b * C_ + c)) * HW_ + n_in + lmm];
      Als[lmm * 257 + c] = (fv - mu) * rs * g1[c] + be1[c];
    }
  }
  // 12 output tiles: 2 (M) x 6 (N of 96 padded cols); wave w owns tiles w and w+8
  const int t0 = wid, t1 = wid + 8;
  const int tm0 = t0 / 6, tn0 = t0 % 6;
  const int tm1 = t1 / 6, tn1 = t1 % 6;
  v8f acc0 = {}, acc1 = {};
  for (int k0 = 0; k0 < C_; k0 += 32) {
    __syncthreads();                       // Als ready (1st iter) / Wls reads done (later iters)
    for (int idx = tid; idx < 32 * 96; idx += 256) {
      int kk = idx / 96, j = idx - kk * 96;
      int kr = k0 + kk;
      float w = 0.f;
      if (j < 72)      w = Ww[kr * 72 + j];
      else if (j < 81) w = Wk[kr * 9 + (j - 72)];
      Wls[kk * 97 + j] = w;
    }
    __syncthreads();
    for (int kk = 0; kk < 32; kk += 4) {
      v2f a0; a0.x = Als[(tm0 * 16 + lane16) * 257 + k0 + kk + koff];
              a0.y = Als[(tm0 * 16 + lane16) * 257 + k0 + kk + koff + 1];
      v2f b0; b0.x = Wls[(kk + koff) * 97 + tn0 * 16 + lane16];
              b0.y = Wls[(kk + koff + 1) * 97 + tn0 * 16 + lane16];
      acc0 = __builtin_amdgcn_wmma_f32_16x16x4_f32(false, a0, false, b0, (short)0, acc0, false, false);
      if (t1 < 12) {                       // wave-uniform guard: EXEC stays all-ones
        v2f a1; a1.x = Als[(tm1 * 16 + lane16) * 257 + k0 + kk + koff];
                a1.y = Als[(tm1 * 16 + lane16) * 257 + k0 + kk + koff + 1];
        v2f b1; b1.x = Wls[(kk + koff) * 97 + tn1 * 16 + lane16];
                b1.y = Wls[(kk + koff + 1) * 97 + tn1 * 16 + lane16];
        acc1 = __builtin_amdgcn_wmma_f32_16x16x4_f32(false, a1, false, b1, (short)0, acc1, false, false);
      }
    }
  }
  __syncthreads();
  {
    const int ml0 = tm0 * 16 + ((lane < 16) ? 0 : 8);
    const int nl0 = tn0 * 16 + lane16;
    for (int v = 0; v < 8; ++v) Lls[(ml0 + v) * 96 + nl0] = acc0[v];
    if (t1 < 12) {
      const int ml1 = tm1 * 16 + ((lane < 16) ? 0 : 8);
      const int nl1 = tn1 * 16 + lane16;
      for (int v = 0; v < 8; ++v) Lls[(ml1 + v) * 96 + nl1] = acc1[v];
    }
  }
  __syncthreads();
  {
    const int m = tid >> 3, g = tid & 7;       // 32 pixels x 8 groups = 256 threads
    const size_t mg = (size_t)(m0 + m);
    float lv[P_];
    float mx = -1e30f;
    for (int p = 0; p < P_; ++p) {
      lv[p] = Lls[m * 96 + p * G_ + g] + bw[p * G_ + g];
      mx = fmaxf(mx, lv[p]);
    }
    float s = 0.f;
    for (int p = 0; p < P_; ++p) { lv[p] = expf(lv[p] - mx); s += lv[p]; }
    float inv = 1.f / s;
    for (int p = 0; p < P_; ++p) wts[mg * GP_ + p * G_ + g] = lv[p] * inv;
    if (g == 0) {
      float ax = anchors[mg * 2 + 0];
      float ay = anchors[mg * 2 + 1];
      for (int p = 0; p < P_; ++p) {
        float kx = ax + Lls[m * 96 + 72 + p] + bk[p];
        kp_out[(mg * P_ + p) * 2 + 0] = kx;
        kp_out[(mg * P_ + p) * 2 + 1] = ay;
      }
    }
  }
}

// ---------------- K3: bilinear gather + group-weighted point aggregation ----------------
__global__ __launch_bounds__(256) void k3_agg(
    const float* __restrict__ value, const float* __restrict__ wts,
    const float* __restrict__ kp, float* __restrict__ agg)
{
  const int tid = threadIdx.x;
  const int wid = tid >> 5, lane = tid & 31;
  const size_t m = (size_t)blockIdx.x * 8 + wid;   // one wave per pixel
  const int b = (int)(m / HW_);
  const int c0 = lane * 8;                         // 8 contiguous channels per lane
  const int g = lane >> 2;                         // group = c0/32
  const float* vb  = value + (size_t)b * HW_ * C_;
  const float* kpm = kp + m * P_ * 2;
  const float* wm  = wts + m * GP_;
  float acc[8] = {0.f,0.f,0.f,0.f,0.f,0.f,0.f,0.f};
  for (int p = 0; p < P_; ++p) {
    float x = kpm[p * 2 + 0] * (float)W_ - 0.5f;   // lane-uniform per wave
    float y = kpm[p * 2 + 1] * (float)H_ - 0.5f;
    float x0f = floorf(x), y0f = floorf(y);
    float wx = x - x0f, wy = y - y0f;
    int x0 = (int)x0f, y0 = (int)y0f;
    float wp = wm[p * G_ + g];
    float cw[4] = { wp*(1.f-wx)*(1.f-wy), wp*wx*(1.f-wy), wp*(1.f-wx)*wy, wp*wx*wy };
    int xs[4] = { x0, x0 + 1, x0,     x0 + 1 };
    int ys[4] = { y0, y0,     y0 + 1, y0 + 1 };
    for (int k = 0; k < 4; ++k) {
      int xi = xs[k], yi = ys[k];
      if (xi >= 0 && xi < W_ && yi >= 0 && yi < H_) {   // wave-uniform branch
        const float* pv = vb + ((size_t)yi * W_ + xi) * C_ + c0;
        float4 va  = *(const float4*)(pv);
        float4 vb4 = *(const float4*)(pv + 4);
        float w = cw[k];
        acc[0] += w * va.x;  acc[1] += w * va.y;  acc[2] += w * va.z;  acc[3] += w * va.w;
        acc[4] += w * vb4.x; acc[5] += w * vb4.y; acc[6] += w * vb4.z; acc[7] += w * vb4.w;
      }
    }
  }
  float4 o0 = { acc[0], acc[1], acc[2], acc[3] };
  float4 o1 = { acc[4], acc[5], acc[6], acc[7] };
  *(float4*)(agg + m * C_ + c0)     = o0;
  *(float4*)(agg + m * C_ + c0 + 4) = o1;
}

// ---------------- K4: out = (agg @ Wo + bo) -> NCHW (fp32 WMMA, async double-buffered) -----
__global__ __launch_bounds__(256) void k4_out(
    const float* __restrict__ agg,
    const float* __restrict__ Wo, const float* __restrict__ bo,
    float* __restrict__ out)
{
  __shared__ float Als[2][32 * 36];
  __shared__ float Bls[2][32 * 132];
  __shared__ float T[128 * 33];     // [channel_local][pixel_local] for coalesced NCHW store
  const int tid = threadIdx.x;
  const int m0 = blockIdx.x * 32;
  const int nb = blockIdx.y * 128;
  const int b = m0 / HW_, n_in = m0 % HW_;
  const int wid = tid >> 5, lane = tid & 31;
  const int wave_m = wid >> 2, wave_n = wid & 3;
  const int lane16 = lane & 15;
  const int koff = (lane < 16) ? 0 : 2;
  const int m_lane = wave_m * 16 + lane16;
  v8f acc0 = {}, acc1 = {};

  auto stageA = [&](int buf, int k0) {
#if USE_ASYNC
    // one async b128 per wave: 4 agg rows (4 lane-groups of 8) per instruction
    int mrow = wid * 4 + (lane >> 3);
    int kq   = (lane & 7) * 4;
    const float* gsrc = agg + (size_t)(m0 + mrow) * C_ + k0 + kq;
    float* ldst = &Als[buf][mrow * 36 + kq];
    ASYNC_LOAD_B128(gsrc, ldst);
#else
    int mm  = tid >> 3;
    int kk0 = (tid & 7) * 4;
    float4 a4 = *(const float4*)(agg + (size_t)(m0 + mm) * C_ + k0 + kk0);
    *(float4*)(&Als[buf][mm * 36 + kk0]) = a4;
#endif
  };
  auto stageB = [&](int buf, int k0) {
#if USE_ASYNC
    for (int i = 0; i < 4; ++i) {
      int r = wid + 8 * i;
      const float* gsrc = Wo + (size_t)(k0 + r) * C_ + nb + lane * 4;
      float* ldst = &Bls[buf][r * 132 + lane * 4];
      ASYNC_LOAD_B128(gsrc, ldst);
    }
#else
    int cg = (tid & 31) * 4;
    int r0 = tid >> 5;
    for (int i = 0; i < 4; ++i) {
      int r = r0 + 8 * i;
      float4 w = *(const float4*)(Wo + (size_t)(k0 + r) * C_ + nb + cg);
      *(float4*)(&Bls[buf][r * 132 + cg]) = w;
    }
#endif
  };
  auto compute = [&](int buf) {
    const int ncol = wave_n * 32 + lane16;
    const float* A = &Als[buf][0];
    const float* Bm = &Bls[buf][0];
    for (int kk = 0; kk < 32; kk += 4) {
      v2f a;  a.x  = A[m_lane * 36 + kk + koff];        a.y  = A[m_lane * 36 + kk + koff + 1];
      v2f b0; b0.x = Bm[(kk + koff) * 132 + ncol];      b0.y = Bm[(kk + koff + 1) * 132 + ncol];
      v2f b1; b1.x = Bm[(kk + koff) * 132 + ncol + 16]; b1.y = Bm[(kk + koff + 1) * 132 + ncol + 16];
      acc0 = __builtin_amdgcn_wmma_f32_16x16x4_f32(false, a, false, b0, (short)0, acc0, false, false);
      acc1 = __builtin_amdgcn_wmma_f32_16x16x4_f32(false, a, false, b1, (short)0, acc1, false, false);
    }
  };

  stageA(0, 0);
  stageB(0, 0);
  ASYNC_WAIT();
  __syncthreads();
  for (int c = 0; c < 8; ++c) {
    const int cur = c & 1, nxt = cur ^ 1;
    if (c < 7) { stageA(nxt, (c + 1) * 32); stageB(nxt, (c + 1) * 32); }
    compute(cur);
    ASYNC_WAIT();
    __syncthreads();
  }

  {
    const int nl = wave_n * 32 + lane16;
    const float boa = bo[nb + nl], bob = bo[nb + nl + 16];
    const int rl = wave_m * 16 + ((lane < 16) ? 0 : 8);
    for (int v = 0; v < 8; ++v) {
      T[nl * 33 + rl + v]        = acc0[v] + boa;
      T[(nl + 16) * 33 + rl + v] = acc1[v] + bob;
    }
  }
  __syncthreads();
  {
    const int col = tid & 31;
    const int r0 = tid >> 5;
    for (int i = 0; i < 16; ++i) {
      int nn = r0 + 8 * i;
      out[(size_t)(b * C_ + nb + nn) * HW_ + n_in + col] = T[nn * 33 + col];
    }
  }
}

extern "C" void kernel_launch(void* const* d_in, const int* in_sizes, int n_in,
                              void* d_out, int out_size, void* d_ws, size_t ws_size,
                              hipStream_t stream)
{
  const float* feats1  = (const float*)d_in[0];
  const float* feats2  = (const float*)d_in[1];
  const float* anchors = (const float*)d_in[2];
  const float* n1g = (const float*)d_in[3];
  const float* n1b = (const float*)d_in[4];
  const float* n2g = (const float*)d_in[5];
  const float* n2b = (const float*)d_in[6];
  const float* Wv  = (const float*)d_in[7];
  const float* bv  = (const float*)d_in[8];
  const float* Ww  = (const float*)d_in[9];
  const float* bw  = (const float*)d_in[10];
  const float* Wk  = (const float*)d_in[11];
  const float* bk  = (const float*)d_in[12];
  const float* Wo  = (const float*)d_in[13];
  const float* bo  = (const float*)d_in[14];

  float* out    = (float*)d_out;
  float* kp_out = out + (size_t)B_ * C_ * HW_;   // kp follows out in the flat output

  float* ws    = (float*)d_ws;
  float* value = ws;                               // MTOT*C floats
  float* agg   = value + (size_t)MTOT * C_;        // MTOT*C floats
  float* wts   = agg   + (size_t)MTOT * C_;        // MTOT*GP floats
  float* mean1 = wts   + (size_t)MTOT * GP_;
  float* rstd1 = mean1 + MTOT;
  float* mean2 = rstd1 + MTOT;
  float* rstd2 = mean2 + MTOT;

  k0_stats<<<dim3(MTOT / 32, 2), 256, 0, stream>>>(feats1, feats2, mean1, rstd1, mean2, rstd2);
  k1_value<<<dim3(MTOT / 32, 2), 256, 0, stream>>>(feats2, mean2, rstd2, n2g, n2b, Wv, bv, value);
  k2_wproj<<<dim3(MTOT / 32), 256, 0, stream>>>(feats1, mean1, rstd1, n1g, n1b, Ww, bw, Wk, bk,
                                                anchors, wts, kp_out);
  k3_agg<<<dim3(MTOT / 8), 256, 0, stream>>>(value, wts, kp_out, agg);
  k4_out<<<dim3(MTOT / 32, 2), 256, 0, stream>>>(agg, Wo, bo, out);

  (void)in_sizes; (void)n_in; (void)out_size; (void)ws_size;
}